// EncoderAGNN_70136815943927
// MI455X (gfx1250) — compile-verified
//
#include <hip/hip_runtime.h>
#include <math.h>

#define IN_CH  512
#define OUT_CH 16

typedef __attribute__((ext_vector_type(2))) float v2f;
typedef __attribute__((ext_vector_type(8))) float v8f;

// ---------------------------------------------------------------------------
// h = relu(x @ W + b)  using V_WMMA_F32_16X16X4_F32 (one 16x16 tile per wave)
// ---------------------------------------------------------------------------
__global__ __launch_bounds__(256) void agnn_gemm_relu(
    const float* __restrict__ x, const float* __restrict__ W,
    const float* __restrict__ b, float* __restrict__ h, int n_nodes) {
  __shared__ float Ws[IN_CH * OUT_CH];  // 32 KB, fits WGP LDS easily
  for (int i = threadIdx.x; i < IN_CH * OUT_CH; i += 256) Ws[i] = W[i];
  __syncthreads();

  const int wave = threadIdx.x >> 5;
  const int lane = threadIdx.x & 31;
  const int tile = blockIdx.x * 8 + wave;
  if (tile * 16 >= n_nodes) return;

  const int m    = lane & 15;   // row (A) / col (B,C) within tile
  const int half = lane >> 4;   // K-phase selector per ISA layout

  const float* __restrict__ xrow = x + (size_t)(tile * 16 + m) * IN_CH;

  v8f c = {};
  for (int k = 0; k < IN_CH; k += 4) {
    const int ka = k + 2 * half;             // even -> 8B aligned
    v2f a = *(const v2f*)(xrow + ka);        // A: row m, K = {ka, ka+1}
    v2f bm;
    bm.x = Ws[ka * OUT_CH + m];              // B: K-row ka,   col m
    bm.y = Ws[(ka + 1) * OUT_CH + m];        // B: K-row ka+1, col m
    c = __builtin_amdgcn_wmma_f32_16x16x4_f32(
        /*neg_a=*/false, a, /*neg_b=*/false, bm,
        /*c_mod=*/(short)0, c, /*reuse_a=*/false, /*reuse_b=*/false);
  }

  const float bias = b[m];
  float* __restrict__ hb = h + (size_t)tile * 16 * OUT_CH;
#pragma unroll
  for (int j = 0; j < 8; ++j) {
    const int row = j + 8 * half;            // C/D layout: VGPR j -> M=j(+8)
    hb[row * OUT_CH + m] = fmaxf(c[j] + bias, 0.0f);
  }
}

// ---------------------------------------------------------------------------
// out[i] = in[i] / max(||in[i]||_2, eps)  -- 16 lanes per node (wave32 shfl)
// ---------------------------------------------------------------------------
__global__ __launch_bounds__(256) void agnn_l2norm(
    const float* __restrict__ in, float* __restrict__ out, int n_nodes,
    float eps) {
  const long long t = (long long)blockIdx.x * blockDim.x + threadIdx.x;
  if (t >= (long long)n_nodes * OUT_CH) return;
  const float v = in[t];
  float sq = v * v;
  sq += __shfl_xor(sq, 1, 16);
  sq += __shfl_xor(sq, 2, 16);
  sq += __shfl_xor(sq, 4, 16);
  sq += __shfl_xor(sq, 8, 16);
  out[t] = v / fmaxf(sqrtf(sq), eps);
}

// ---------------------------------------------------------------------------
// Pass A: e = exp(beta * <xn[dst], xn[src]>); s[dst] += e   (16 lanes/edge)
// Softmax max-shift elided: logits bounded by |beta| (cosine similarity).
// ---------------------------------------------------------------------------
__global__ __launch_bounds__(256) void agnn_edge_softmax(
    const int* __restrict__ src_idx, const int* __restrict__ dst_idx,
    const float* __restrict__ xn, float* __restrict__ ebuf,
    float* __restrict__ s, const float* __restrict__ beta_p, long long E_edges,
    int n_nodes) {
  const long long t = (long long)blockIdx.x * blockDim.x + threadIdx.x;
  const long long edge = t >> 4;
  const int c = (int)(t & 15);
  const long long total = E_edges + n_nodes;
  if (edge >= total) return;

  int sidx, didx;
  if (edge < E_edges) {
    sidx = src_idx[edge];
    didx = dst_idx[edge];
  } else {
    sidx = didx = (int)(edge - E_edges);  // self-loop
  }

  float p = xn[(size_t)sidx * OUT_CH + c] * xn[(size_t)didx * OUT_CH + c];
  p += __shfl_xor(p, 1, 16);
  p += __shfl_xor(p, 2, 16);
  p += __shfl_xor(p, 4, 16);
  p += __shfl_xor(p, 8, 16);

  if (c == 0) {
    const float e = expf(beta_p[0] * p);
    ebuf[edge] = e;
    atomicAdd(&s[didx], e);
  }
}

// ---------------------------------------------------------------------------
// Pass B: out[dst] += (e[edge] / s[dst]) * feat[src]   (16 lanes/edge)
// ---------------------------------------------------------------------------
__global__ __launch_bounds__(256) void agnn_edge_aggregate(
    const int* __restrict__ src_idx, const int* __restrict__ dst_idx,
    const float* __restrict__ feat, const float* __restrict__ ebuf,
    const float* __restrict__ s, float* __restrict__ out, long long E_edges,
    int n_nodes) {
  const long long t = (long long)blockIdx.x * blockDim.x + threadIdx.x;
  const long long edge = t >> 4;
  const int c = (int)(t & 15);
  const long long total = E_edges + n_nodes;
  if (edge >= total) return;

  int sidx, didx;
  if (edge < E_edges) {
    sidx = src_idx[edge];
    didx = dst_idx[edge];
  } else {
    sidx = didx = (int)(edge - E_edges);
  }

  const float alpha = ebuf[edge] / s[didx];  // same addr across lanes -> 1 req
  atomicAdd(&out[(size_t)didx * OUT_CH + c],
            alpha * feat[(size_t)sidx * OUT_CH + c]);
}

// ---------------------------------------------------------------------------
extern "C" void kernel_launch(void* const* d_in, const int* in_sizes, int n_in,
                              void* d_out, int out_size, void* d_ws,
                              size_t ws_size, hipStream_t stream) {
  const float* x     = (const float*)d_in[0];
  const int*   ei    = (const int*)d_in[1];  // [2, E] int32
  const float* W     = (const float*)d_in[2];
  const float* b     = (const float*)d_in[3];
  const float* beta1 = (const float*)d_in[4];
  const float* beta2 = (const float*)d_in[5];

  const int       N = in_sizes[0] / IN_CH;       // 100000
  const long long E = (long long)in_sizes[1] / 2;  // 3200000
  const long long T = E + N;                     // edges incl. self-loops

  const int* src = ei;
  const int* dst = ei + E;

  // Workspace layout (floats)
  float* h0 = (float*)d_ws;                  // N*16
  float* xn = h0 + (size_t)N * OUT_CH;       // N*16
  float* h1 = xn + (size_t)N * OUT_CH;       // N*16
  float* h2 = h1 + (size_t)N * OUT_CH;       // N*16
  float* s  = h2 + (size_t)N * OUT_CH;       // N
  float* eb = s + N;                         // E+N

  const int tiles = (N + 15) / 16;
  const int gemm_blocks = (tiles + 7) / 8;   // 8 waves (tiles) per block
  const long long node_thr = (long long)N * OUT_CH;
  const int node_blocks = (int)((node_thr + 255) / 256);
  const long long edge_thr = T * 16;
  const int edge_blocks = (int)((edge_thr + 255) / 256);

  // 1) h0 = relu(x @ W + b)  (WMMA f32 16x16x4)
  agnn_gemm_relu<<<gemm_blocks, 256, 0, stream>>>(x, W, b, h0, N);

  // ---- layer 1 ----
  agnn_l2norm<<<node_blocks, 256, 0, stream>>>(h0, xn, N, 1e-12f);
  hipMemsetAsync(s, 0, (size_t)N * sizeof(float), stream);
  hipMemsetAsync(h1, 0, (size_t)N * OUT_CH * sizeof(float), stream);
  agnn_edge_softmax<<<edge_blocks, 256, 0, stream>>>(src, dst, xn, eb, s,
                                                     beta1, E, N);
  agnn_edge_aggregate<<<edge_blocks, 256, 0, stream>>>(src, dst, h0, eb, s, h1,
                                                       E, N);

  // ---- layer 2 ----
  agnn_l2norm<<<node_blocks, 256, 0, stream>>>(h1, xn, N, 1e-12f);
  hipMemsetAsync(s, 0, (size_t)N * sizeof(float), stream);
  hipMemsetAsync(h2, 0, (size_t)N * OUT_CH * sizeof(float), stream);
  agnn_edge_softmax<<<edge_blocks, 256, 0, stream>>>(src, dst, xn, eb, s,
                                                     beta2, E, N);
  agnn_edge_aggregate<<<edge_blocks, 256, 0, stream>>>(src, dst, h1, eb, s, h2,
                                                       E, N);

  // final normalize (eps = 0.0005) straight into d_out
  agnn_l2norm<<<node_blocks, 256, 0, stream>>>(h2, (float*)d_out, N, 5e-4f);
}